// PEFeat_61211873903151
// MI455X (gfx1250) — compile-verified
//
#include <hip/hip_runtime.h>
#include <hip/hip_bf16.h>

typedef float v2f __attribute__((ext_vector_type(2)));
typedef float v8f __attribute__((ext_vector_type(8)));

// D = A(16x4) * B(4x16) + C(16x16), all fp32, wave32.
// 8-arg form: (neg_a, A, neg_b, B, c_mod, C, reuse_a, reuse_b)
__device__ __forceinline__ v8f wmma_f32(v2f a, v2f b, v8f c) {
    return __builtin_amdgcn_wmma_f32_16x16x4_f32(false, a, false, b, (short)0, c, false, false);
}

// ---------------------------------------------------------------------------
// Layer-1 convs (Cin=3 / Cin=32): memory bound, scalar kernel.
// ---------------------------------------------------------------------------
__global__ void conv_small_relu(const float* __restrict__ in, const float* __restrict__ w,
                                const float* __restrict__ bias, float* __restrict__ out,
                                int Cin, int Cout, int Npts) {
    int n = blockIdx.x * blockDim.x + threadIdx.x;
    int b = blockIdx.z;
    if (n >= Npts) return;
    const float* inb = in + (size_t)b * Cin * Npts;
    float* outb      = out + (size_t)b * Cout * Npts;
    float xin[32];
    for (int i = 0; i < Cin; ++i) xin[i] = inb[(size_t)i * Npts + n];
    for (int o = 0; o < Cout; ++o) {
        float acc = bias[o];
        for (int i = 0; i < Cin; ++i) acc += w[o * Cin + i] * xin[i];
        outb[(size_t)o * Npts + n] = fmaxf(acc, 0.f);
    }
}

// ---------------------------------------------------------------------------
// Generic conv1x1 GEMM via f32 WMMA. out[b][o][n] = relu(sum_i w[o][i]*in[b][i][n] + bias[o])
// Optional fused mean-pool epilogue (do_pool): accumulate per-(b,o) row sums
// into pool_out via atomics instead of storing the [B,1024,N] activation.
// One wave = one 16(o) x 16(n) tile; 8 waves/block tile n.
// ---------------------------------------------------------------------------
__global__ void __launch_bounds__(256)
conv_wmma_relu(const float* __restrict__ in, const float* __restrict__ w,
               const float* __restrict__ bias, float* __restrict__ out,
               int Cin, int Cout, int Npts,
               long in_bstride, long out_bstride,
               int do_pool, float* __restrict__ pool_out) {
    int wave = threadIdx.x >> 5;
    int lane = threadIdx.x & 31;
    int lo16 = lane & 15;
    bool hi  = lane >= 16;
    int n0 = (blockIdx.x * 8 + wave) * 16;
    int o0 = blockIdx.y * 16;
    int b  = blockIdx.z;
    if (n0 >= Npts) return;                 // wave-uniform: EXEC stays full
    const float* inb = in + (size_t)b * in_bstride;

    v8f acc = {};
    int arow = o0 + lo16;
    int ncol = n0 + lo16;
    for (int k0 = 0; k0 < Cin; k0 += 4) {
        int ka = k0 + (hi ? 2 : 0);
        v2f a, bf;
        a.x  = w[(size_t)arow * Cin + ka];
        a.y  = w[(size_t)arow * Cin + ka + 1];
        bf.x = inb[(size_t)ka * Npts + ncol];
        bf.y = inb[(size_t)(ka + 1) * Npts + ncol];
        acc = wmma_f32(a, bf, acc);
    }

#pragma unroll
    for (int r = 0; r < 8; ++r) {
        int orow = o0 + r + (hi ? 8 : 0);
        float v = acc[r] + bias[orow];
        v = fmaxf(v, 0.f);
        if (do_pool) {
            float s = v;
            s += __shfl_xor(s, 1, 32);
            s += __shfl_xor(s, 2, 32);
            s += __shfl_xor(s, 4, 32);
            s += __shfl_xor(s, 8, 32);      // sum over the 16 n-columns of this row
            if (lo16 == 0) atomicAdd(pool_out + (size_t)b * Cout + orow, s);
        } else {
            out[(size_t)b * out_bstride + (size_t)orow * Npts + ncol] = v;
        }
    }
}

// ---------------------------------------------------------------------------
// Pass A of cb_layer: rowmax/rowsum of energy[row, col] = sum_c Y[c,row]*X[c,col]
// (online softmax over cols). One wave owns 16 rows; Y A-fragments are
// loop-invariant and preloaded into registers.
// ---------------------------------------------------------------------------
template <int C>
__global__ void __launch_bounds__(256)
rowstats_kernel(const float* __restrict__ Y, const float* __restrict__ X,
                float* __restrict__ rmax, float* __restrict__ rsum, int Npts) {
    int wave = threadIdx.x >> 5;
    int lane = threadIdx.x & 31;
    int lo16 = lane & 15;
    bool hi  = lane >= 16;
    int n0 = (blockIdx.x * 8 + wave) * 16;
    int b  = blockIdx.y;
    const float* Yb = Y + (size_t)b * C * Npts;
    const float* Xb = X + (size_t)b * C * Npts;

    v2f afr[C / 4];
#pragma unroll
    for (int j = 0; j < C / 4; ++j) {
        int k = 4 * j + (hi ? 2 : 0);
        afr[j].x = Yb[(size_t)k * Npts + n0 + lo16];
        afr[j].y = Yb[(size_t)(k + 1) * Npts + n0 + lo16];
    }

    float m_run[8], s_run[8];
#pragma unroll
    for (int r = 0; r < 8; ++r) { m_run[r] = -3.0e38f; s_run[r] = 0.f; }

    for (int m0 = 0; m0 < Npts; m0 += 16) {
        v8f e = {};
#pragma unroll
        for (int j = 0; j < C / 4; ++j) {
            int k = 4 * j + (hi ? 2 : 0);
            v2f bf;
            bf.x = Xb[(size_t)k * Npts + m0 + lo16];
            bf.y = Xb[(size_t)(k + 1) * Npts + m0 + lo16];
            e = wmma_f32(afr[j], bf, e);
        }
#pragma unroll
        for (int r = 0; r < 8; ++r) {       // online softmax update (per row slot)
            float v  = e[r];
            float mn = fmaxf(m_run[r], v);
            s_run[r] = s_run[r] * __expf(m_run[r] - mn) + __expf(v - mn);
            m_run[r] = mn;
        }
    }
    // merge across the 16 lanes of each half-wave (halves hold disjoint rows)
#pragma unroll
    for (int r = 0; r < 8; ++r) {
        float m = m_run[r], s = s_run[r];
        for (int off = 1; off < 16; off <<= 1) {
            float om = __shfl_xor(m, off, 32);
            float os = __shfl_xor(s, off, 32);
            float mn = fmaxf(m, om);
            s = s * __expf(m - mn) + os * __expf(om - mn);
            m = mn;
        }
        m_run[r] = m; s_run[r] = s;
    }
    if (lo16 == 0) {
#pragma unroll
        for (int r = 0; r < 8; ++r) {
            int row = n0 + r + (hi ? 8 : 0);
            rmax[(size_t)b * Npts + row] = m_run[r];
            rsum[(size_t)b * Npts + row] = s_run[r];
        }
    }
}

// ---------------------------------------------------------------------------
// Passes B/C of cb_layer, fused: for a fixed 16-column tile, stream all rows,
// recompute the energy tile (WMMA), form weight = exp(e - rm)/rs, and fold it
// straight into a second WMMA chain:  D[M=col, N=c] += W^T x V^T.
// MODE 0 ("B"): rows normalized (rm/rs indexed by row), writes colsum,
//               out = resid + acc/(1e-9+colsum[col]).   (y_r branch)
// MODE 1 ("C"): cols normalized (rm/rs indexed by col), V is pre-scaled by
//               1/(1e-9+colsum), out = resid + acc.     (x_r branch)
// Weight tile is moved from WMMA C-layout to A-operand layout with
// shfl_xor(16) half-wave exchanges — no LDS round trip.
// ---------------------------------------------------------------------------
template <int C, int MODE>
__global__ void __launch_bounds__(128)
attn_apply(const float* __restrict__ Rsrc,   // energy rows source  [B][C][N]
           const float* __restrict__ Csrc,   // energy cols source  [B][C][N]
           const float* __restrict__ V,      // values              [B][C][N]
           const float* __restrict__ resid,  // residual            [B][C][N]
           float* __restrict__ out, long out_bstride,
           const float* __restrict__ rmax, const float* __restrict__ rsum,
           float* __restrict__ colsum, int Npts) {
    int wave = threadIdx.x >> 5;
    int lane = threadIdx.x & 31;
    int lo16 = lane & 15;
    bool hi  = lane >= 16;
    int col0 = (blockIdx.x * 4 + wave) * 16;
    int b    = blockIdx.y;
    if (col0 >= Npts) return;
    const float* Rb = Rsrc + (size_t)b * C * Npts;
    const float* Cb = Csrc + (size_t)b * C * Npts;
    const float* Vb = V    + (size_t)b * C * Npts;

    // loop-invariant column-side energy B-fragments
    v2f cf[C / 4];
#pragma unroll
    for (int j = 0; j < C / 4; ++j) {
        int k = 4 * j + (hi ? 2 : 0);
        cf[j].x = Cb[(size_t)k * Npts + col0 + lo16];
        cf[j].y = Cb[(size_t)(k + 1) * Npts + col0 + lo16];
    }
    float col_rm = 0.f, col_rsi = 0.f;
    if (MODE == 1) {
        col_rm  = rmax[(size_t)b * Npts + col0 + lo16];
        col_rsi = 1.0f / rsum[(size_t)b * Npts + col0 + lo16];
    }

    v8f dacc[C / 16];
    v8f vzero = {};
#pragma unroll
    for (int t = 0; t < C / 16; ++t) dacc[t] = vzero;
    float cs_lane = 0.f;

    for (int row0 = 0; row0 < Npts; row0 += 16) {
        // energy tile (rows from Rsrc, cols from Csrc)
        v8f e = {};
#pragma unroll
        for (int j = 0; j < C / 4; ++j) {
            int k = 4 * j + (hi ? 2 : 0);
            v2f af;
            af.x = Rb[(size_t)k * Npts + row0 + lo16];
            af.y = Rb[(size_t)(k + 1) * Npts + row0 + lo16];
            e = wmma_f32(af, cf[j], e);
        }
        // softmax weights
        float w[8];
        if (MODE == 0) {
#pragma unroll
            for (int r = 0; r < 8; ++r) {
                int row  = row0 + r + (hi ? 8 : 0);
                float rm = rmax[(size_t)b * Npts + row];
                float ri = 1.0f / rsum[(size_t)b * Npts + row];
                w[r] = __expf(e[r] - rm) * ri;
                cs_lane += w[r];
            }
        } else {
#pragma unroll
            for (int r = 0; r < 8; ++r) w[r] = __expf(e[r] - col_rm) * col_rsi;
        }
        // apply GEMM: D[M=col, N=c] += W^T(16x16, K-chunks of 4) x V^T
#pragma unroll
        for (int jj = 0; jj < 4; ++jj) {
            int k0 = 4 * jj;
            int sA = (jj < 2) ? ((k0 + 2) & 7) : (k0 & 7);
            float sx_a = __shfl_xor(w[sA], 16, 32);
            float sx_b = __shfl_xor(w[sA + 1], 16, 32);
            v2f af;
            if (jj < 2) { af.x = hi ? sx_a : w[k0 & 7];       af.y = hi ? sx_b : w[(k0 + 1) & 7]; }
            else        { af.x = hi ? w[(k0 + 2) & 7] : sx_a; af.y = hi ? w[(k0 + 3) & 7] : sx_b; }
            int krow = row0 + k0 + (hi ? 2 : 0);
#pragma unroll
            for (int t = 0; t < C / 16; ++t) {
                v2f bf;
                bf.x = Vb[(size_t)(16 * t + lo16) * Npts + krow];
                bf.y = Vb[(size_t)(16 * t + lo16) * Npts + krow + 1];
                dacc[t] = wmma_f32(af, bf, dacc[t]);
            }
        }
    }

    float cs_tot = 0.f;
    if (MODE == 0) {
        cs_tot = cs_lane + __shfl_xor(cs_lane, 16, 32);
        if (!hi) colsum[(size_t)b * Npts + col0 + lo16] = cs_tot;
    }
    float* outb       = out   + (size_t)b * out_bstride;
    const float* resb = resid + (size_t)b * C * Npts;
#pragma unroll
    for (int t = 0; t < C / 16; ++t) {
#pragma unroll
        for (int r = 0; r < 8; ++r) {
            int col = col0 + r + (hi ? 8 : 0);
            int c   = 16 * t + lo16;
            float v = dacc[t][r];
            if (MODE == 0) {
                float cs = __shfl(cs_tot, r + (hi ? 8 : 0), 16);
                v = v / (1e-9f + cs);
            }
            v += resb[(size_t)c * Npts + col];
            outb[(size_t)c * Npts + col] = v;
        }
    }
}

// xt[c][n] = x[c][n] / (1e-9 + colsum[n])
__global__ void prescale_kernel(const float* __restrict__ x, const float* __restrict__ colsum,
                                float* __restrict__ xt, int C, int Npts) {
    int n = blockIdx.x * blockDim.x + threadIdx.x;
    int c = blockIdx.y, b = blockIdx.z;
    if (n >= Npts) return;
    size_t idx = ((size_t)b * C + c) * Npts + n;
    xt[idx] = x[idx] / (1e-9f + colsum[(size_t)b * Npts + n]);
}

// broadcast mean-pooled features across N
__global__ void pool_bcast_kernel(const float* __restrict__ aps, float* __restrict__ out,
                                  long out_bstride, int Cout, int Npts, float inv_n) {
    int n = blockIdx.x * blockDim.x + threadIdx.x;
    int c = blockIdx.y, b = blockIdx.z;
    if (n >= Npts) return;
    out[(size_t)b * out_bstride + (size_t)c * Npts + n] = aps[(size_t)b * Cout + c] * inv_n;
}

// ---------------------------------------------------------------------------
extern "C" void kernel_launch(void* const* d_in, const int* in_sizes, int n_in,
                              void* d_out, int out_size, void* d_ws, size_t ws_size,
                              hipStream_t stream) {
    (void)in_sizes; (void)n_in; (void)out_size; (void)ws_size;
    const int B = 8, N = 4096;
    const float* x   = (const float*)d_in[0];
    const float* emb = (const float*)d_in[1];
    const float* w1  = (const float*)d_in[2];
    const float* b1  = (const float*)d_in[3];
    const float* w2  = (const float*)d_in[4];
    const float* b2  = (const float*)d_in[5];
    const float* ew1 = (const float*)d_in[6];
    const float* eb1 = (const float*)d_in[7];
    const float* ew2 = (const float*)d_in[8];
    const float* eb2 = (const float*)d_in[9];
    const float* w5  = (const float*)d_in[10];
    const float* b5  = (const float*)d_in[11];
    const float* w6  = (const float*)d_in[12];
    const float* b6  = (const float*)d_in[13];
    float* out = (float*)d_out;

    float* f   = (float*)d_ws;                 // ~135 MB total
    float* h1  = f; f += (size_t)B * 64 * N;
    float* e1  = f; f += (size_t)B * 64 * N;
    float* h2  = f; f += (size_t)B * 128 * N;
    float* e2  = f; f += (size_t)B * 128 * N;
    float* vt  = f; f += (size_t)B * 128 * N;  // prescaled values buffer
    float* rmx = f; f += (size_t)B * N;
    float* rsm = f; f += (size_t)B * N;
    float* csm = f; f += (size_t)B * N;
    float* g1  = f; f += (size_t)B * 512 * N;
    float* aps = f; f += (size_t)B * 1024;
    const long OBS = 1408L * N;                // output batch stride

    dim3 blk(256);
    // layer 1
    conv_small_relu<<<dim3(N / 256, 1, B), blk, 0, stream>>>(x,   w1,  b1,  h1, 3,  64, N);
    conv_small_relu<<<dim3(N / 256, 1, B), blk, 0, stream>>>(emb, ew1, eb1, e1, 32, 64, N);
    // cb_layer 1 (x=h1, y=e1) -> out rows [0,128)
    rowstats_kernel<64><<<dim3(N / 128, B), 256, 0, stream>>>(e1, h1, rmx, rsm, N);
    attn_apply<64, 0><<<dim3(N / 64, B), 128, 0, stream>>>(e1, h1, e1, h1, out, OBS, rmx, rsm, csm, N);
    prescale_kernel<<<dim3(N / 256, 64, B), blk, 0, stream>>>(h1, csm, vt, 64, N);
    attn_apply<64, 1><<<dim3(N / 64, B), 128, 0, stream>>>(h1, e1, vt, e1, out + 64L * N, OBS, rmx, rsm, csm, N);
    // layer 2 convs
    conv_wmma_relu<<<dim3(N / 128, 128 / 16, B), 256, 0, stream>>>(h1, w2,  b2,  h2, 64, 128, N, 64L * N, 128L * N, 0, nullptr);
    conv_wmma_relu<<<dim3(N / 128, 128 / 16, B), 256, 0, stream>>>(e1, ew2, eb2, e2, 64, 128, N, 64L * N, 128L * N, 0, nullptr);
    // cb_layer 2 (x=h2, y=e2) -> out rows [128,384)
    rowstats_kernel<128><<<dim3(N / 128, B), 256, 0, stream>>>(e2, h2, rmx, rsm, N);
    attn_apply<128, 0><<<dim3(N / 64, B), 128, 0, stream>>>(e2, h2, e2, h2, out + 128L * N, OBS, rmx, rsm, csm, N);
    prescale_kernel<<<dim3(N / 256, 128, B), blk, 0, stream>>>(h2, csm, vt, 128, N);
    attn_apply<128, 1><<<dim3(N / 64, B), 128, 0, stream>>>(h2, e2, vt, e2, out + 256L * N, OBS, rmx, rsm, csm, N);
    // head: 256 -> 512 -> 1024 (+ fused mean pool), broadcast into rows [384,1408)
    conv_wmma_relu<<<dim3(N / 128, 512 / 16, B), 256, 0, stream>>>(out + 128L * N, w5, b5, g1, 256, 512, N, OBS, 512L * N, 0, nullptr);
    hipMemsetAsync(aps, 0, (size_t)B * 1024 * sizeof(float), stream);
    conv_wmma_relu<<<dim3(N / 128, 1024 / 16, B), 256, 0, stream>>>(g1, w6, b6, nullptr, 512, 1024, N, 512L * N, 0, 1, aps);
    pool_bcast_kernel<<<dim3(N / 256, 1024, B), blk, 0, stream>>>(aps, out + 384L * N, OBS, 1024, N, 1.0f / N);
}